// MatchHistogram_22737556865702
// MI455X (gfx1250) — compile-verified
//
#include <hip/hip_runtime.h>
#include <math.h>

typedef __attribute__((ext_vector_type(2))) float v2f;
typedef __attribute__((ext_vector_type(4))) float v4f;
typedef __attribute__((ext_vector_type(8))) float v8f;

#define NPIX   (2048*2048)      // pixels per channel
#define EPS    1e-5f
#define NB_RED 256              // blocks in reduction pass
#define NT_RED 256              // threads per reduction block
#define WS_RESULT_OFF (NB_RED*18)   // float offset in ws for the 3x4 [T|b] matrix

// ---------------------------------------------------------------------------
// Pass 1: per-channel sums and cross-moments for both images (18 quantities).
// float4 (b128) loads; deterministic LDS tree reduction; per-block partials.
// ---------------------------------------------------------------------------
__global__ void mh_reduce(const float* __restrict__ tgt,
                          const float* __restrict__ src,
                          float* __restrict__ ws) {
  float a[18];
#pragma unroll
  for (int i = 0; i < 18; ++i) a[i] = 0.f;

  const int tid    = blockIdx.x * blockDim.x + threadIdx.x;
  const int stride = gridDim.x * blockDim.x;
  const int nquads = NPIX / 4;
  for (int q = tid; q < nquads; q += stride) {
    const size_t p = (size_t)q * 4;
    v4f t0 = *(const v4f*)(tgt + p);
    v4f t1 = *(const v4f*)(tgt + NPIX + p);
    v4f t2 = *(const v4f*)(tgt + 2 * NPIX + p);
    v4f s0 = *(const v4f*)(src + p);
    v4f s1 = *(const v4f*)(src + NPIX + p);
    v4f s2 = *(const v4f*)(src + 2 * NPIX + p);
#pragma unroll
    for (int j = 0; j < 4; ++j) {
      float x0 = t0[j], x1 = t1[j], x2 = t2[j];
      float y0 = s0[j], y1 = s1[j], y2 = s2[j];
      a[0] += x0;      a[1] += x1;      a[2] += x2;
      a[3] += x0 * x0; a[4] += x0 * x1; a[5] += x0 * x2;
      a[6] += x1 * x1; a[7] += x1 * x2; a[8] += x2 * x2;
      a[9]  += y0;      a[10] += y1;      a[11] += y2;
      a[12] += y0 * y0; a[13] += y0 * y1; a[14] += y0 * y2;
      a[15] += y1 * y1; a[16] += y1 * y2; a[17] += y2 * y2;
    }
  }

  __shared__ float red[18 * NT_RED];
  const int t = threadIdx.x;
#pragma unroll
  for (int q = 0; q < 18; ++q) red[q * NT_RED + t] = a[q];
  __syncthreads();
  for (int s = NT_RED / 2; s > 0; s >>= 1) {
    if (t < s) {
#pragma unroll
      for (int q = 0; q < 18; ++q)
        red[q * NT_RED + t] += red[q * NT_RED + t + s];
    }
    __syncthreads();
  }
  if (t == 0) {
#pragma unroll
    for (int q = 0; q < 18; ++q) ws[blockIdx.x * 18 + q] = red[q * NT_RED];
  }
}

// ---------------------------------------------------------------------------
// Pass 2: tiny 3x3 linear algebra on one thread.
// ---------------------------------------------------------------------------
__device__ inline void jacobi3(float* A, float* V) {
  for (int i = 0; i < 9; ++i) V[i] = 0.f;
  V[0] = V[4] = V[8] = 1.f;
  for (int sweep = 0; sweep < 16; ++sweep) {
    for (int pair = 0; pair < 3; ++pair) {
      const int p = (pair == 2) ? 1 : 0;
      const int q = (pair == 0) ? 1 : 2;
      float apq = A[p * 3 + q];
      if (fabsf(apq) > 1e-30f) {
        float app = A[p * 3 + p], aqq = A[q * 3 + q];
        float tau = (aqq - app) / (2.0f * apq);
        float tt  = ((tau >= 0.f) ? 1.0f : -1.0f) /
                    (fabsf(tau) + sqrtf(1.0f + tau * tau));
        float c = 1.0f / sqrtf(1.0f + tt * tt);
        float s = tt * c;
        for (int k = 0; k < 3; ++k) {            // A = A * J
          float akp = A[k * 3 + p], akq = A[k * 3 + q];
          A[k * 3 + p] = c * akp - s * akq;
          A[k * 3 + q] = s * akp + c * akq;
        }
        for (int k = 0; k < 3; ++k) {            // A = J^T * A
          float apk = A[p * 3 + k], aqk = A[q * 3 + k];
          A[p * 3 + k] = c * apk - s * aqk;
          A[q * 3 + k] = s * apk + c * aqk;
        }
        for (int k = 0; k < 3; ++k) {            // V = V * J
          float vkp = V[k * 3 + p], vkq = V[k * 3 + q];
          V[k * 3 + p] = c * vkp - s * vkq;
          V[k * 3 + q] = s * vkp + c * vkq;
        }
      }
    }
  }
}

__device__ inline void pca3(float* C, float* P) {   // P = V sqrt(L) V^T
  float V[9];
  jacobi3(C, V);           // C -> ~diag(eigenvalues)
  float e[3];
  for (int i = 0; i < 3; ++i) {
    float s = sqrtf(C[i * 3 + i]);
    e[i] = (s == s) ? s : 0.f;   // nan -> 0 (matches reference)
  }
  for (int i = 0; i < 3; ++i)
    for (int j = 0; j < 3; ++j) {
      float s = 0.f;
      for (int k = 0; k < 3; ++k) s += V[i * 3 + k] * e[k] * V[j * 3 + k];
      P[i * 3 + j] = s;
    }
}

__device__ inline void inv3(const float* A, float* B) {
  float a = A[0], b = A[1], c = A[2], d = A[3], e = A[4],
        f = A[5], g = A[6], h = A[7], i = A[8];
  float det = a * (e * i - f * h) - b * (d * i - f * g) + c * (d * h - e * g);
  float id = 1.0f / det;
  B[0] = (e * i - f * h) * id; B[1] = (c * h - b * i) * id; B[2] = (b * f - c * e) * id;
  B[3] = (f * g - d * i) * id; B[4] = (a * i - c * g) * id; B[5] = (c * d - a * f) * id;
  B[6] = (d * h - e * g) * id; B[7] = (b * g - a * h) * id; B[8] = (a * e - b * d) * id;
}

__global__ void mh_finalize(float* __restrict__ ws) {
  __shared__ float tot[18];
  const int t = threadIdx.x;
  if (t < 18) {
    float s = 0.f;
    for (int b = 0; b < NB_RED; ++b) s += ws[b * 18 + t];
    tot[t] = s;
  }
  __syncthreads();
  if (t == 0) {
    const float invN = 1.0f / (float)NPIX;
    float mt[3] = { tot[0] * invN, tot[1] * invN, tot[2] * invN };
    float ms[3] = { tot[9] * invN, tot[10] * invN, tot[11] * invN };

    float Ct[9], Cs[9];
    Ct[0] = tot[3] * invN - mt[0] * mt[0] + EPS;
    Ct[1] = Ct[3] = tot[4] * invN - mt[0] * mt[1];
    Ct[2] = Ct[6] = tot[5] * invN - mt[0] * mt[2];
    Ct[4] = tot[6] * invN - mt[1] * mt[1] + EPS;
    Ct[5] = Ct[7] = tot[7] * invN - mt[1] * mt[2];
    Ct[8] = tot[8] * invN - mt[2] * mt[2] + EPS;

    Cs[0] = tot[12] * invN - ms[0] * ms[0] + EPS;
    Cs[1] = Cs[3] = tot[13] * invN - ms[0] * ms[1];
    Cs[2] = Cs[6] = tot[14] * invN - ms[0] * ms[2];
    Cs[4] = tot[15] * invN - ms[1] * ms[1] + EPS;
    Cs[5] = Cs[7] = tot[16] * invN - ms[1] * ms[2];
    Cs[8] = tot[17] * invN - ms[2] * ms[2] + EPS;

    float Pt[9], Ps[9], Pti[9], T[9];
    pca3(Ct, Pt);
    pca3(Cs, Ps);
    inv3(Pt, Pti);
    for (int i = 0; i < 3; ++i)
      for (int j = 0; j < 3; ++j) {
        float s = 0.f;
        for (int k = 0; k < 3; ++k) s += Ps[i * 3 + k] * Pti[k * 3 + j];
        T[i * 3 + j] = s;
      }
    // Fold both means into a bias: out = T*x + b, b = ms - T*mt
    for (int c = 0; c < 3; ++c) {
      float b = ms[c] - (T[c * 3 + 0] * mt[0] + T[c * 3 + 1] * mt[1] + T[c * 3 + 2] * mt[2]);
      ws[WS_RESULT_OFF + c * 4 + 0] = T[c * 3 + 0];
      ws[WS_RESULT_OFF + c * 4 + 1] = T[c * 3 + 1];
      ws[WS_RESULT_OFF + c * 4 + 2] = T[c * 3 + 2];
      ws[WS_RESULT_OFF + c * 4 + 3] = b;
    }
  }
}

// ---------------------------------------------------------------------------
// Pass 3: out(3xN) = [T|b](3x4) x [x0;x1;x2;1](4xN) via V_WMMA_F32_16X16X4_F32.
// Each wave iteration handles a 64-pixel group with b128 memory ops:
//   lane n loads float4 = pixels {base+4n .. base+4n+3} of its channel
//   WMMA j (j=0..3) computes columns = pixels {base+4n+j : n=0..15}
//   lane n then holds contiguous float4 of each output channel -> b128 stores.
// A (16x4): lane = M (rows 0-2 = [T|b] rows); (half,j) = K.
// B (4x16): lane&15 = N, (half,j) = K: K0=ch0, K1=ch1, K2=ch2, K3=ones.
// Loop trip count is an SGPR (readfirstlane) so EXEC is all-1s at every WMMA.
// ---------------------------------------------------------------------------
__global__ void mh_apply_wmma(const float* __restrict__ in,
                              float* __restrict__ out,
                              const float* __restrict__ M) {
  const int lane = threadIdx.x & 31;
  const int n    = lane & 15;       // matrix column (pixel slot) / A row
  const int half = lane >> 4;       // selects K pair {0,1} vs {2,3}

  float a0 = 0.f, a1 = 0.f;
  if (n < 3) {                      // rows 3..15 of A are zero
    a0 = M[n * 4 + 2 * half + 0];
    a1 = M[n * 4 + 2 * half + 1];
  }
  const v2f A = { a0, a1 };

  const int wave    = __builtin_amdgcn_readfirstlane(
                        (blockIdx.x * blockDim.x + threadIdx.x) >> 5);
  const int nwaves  = (gridDim.x * blockDim.x) >> 5;
  const int ngroups = NPIX / 64;    // 65536, exact

  for (int g = wave; g < ngroups; g += nwaves) {
    const size_t p4 = (size_t)g * 64 + 4 * n;   // lane's float4 base pixel
    // q0: lanes 0-15 = ch0, lanes 16-31 = ch2 (two contiguous 256B segments)
    v4f q0 = *(const v4f*)(in + (half ? (size_t)2 * NPIX : 0) + p4);
    v4f q1 = { 1.f, 1.f, 1.f, 1.f };            // K3 = ones row on half==1
    if (!half) q1 = *(const v4f*)(in + NPIX + p4);

    v8f C0 = {}, C1 = {}, C2 = {}, C3 = {};
    v2f B0 = { q0[0], q1[0] };
    v2f B1 = { q0[1], q1[1] };
    v2f B2 = { q0[2], q1[2] };
    v2f B3 = { q0[3], q1[3] };
    C0 = __builtin_amdgcn_wmma_f32_16x16x4_f32(false, A, false, B0, (short)0, C0, false, false);
    C1 = __builtin_amdgcn_wmma_f32_16x16x4_f32(false, A, false, B1, (short)0, C1, false, false);
    C2 = __builtin_amdgcn_wmma_f32_16x16x4_f32(false, A, false, B2, (short)0, C2, false, false);
    C3 = __builtin_amdgcn_wmma_f32_16x16x4_f32(false, A, false, B3, (short)0, C3, false, false);

    if (!half) {                    // lanes 0-15 hold rows M=0..2 in C[0..2]
      v4f o0 = { C0[0], C1[0], C2[0], C3[0] };
      v4f o1 = { C0[1], C1[1], C2[1], C3[1] };
      v4f o2 = { C0[2], C1[2], C2[2], C3[2] };
      *(v4f*)(out + p4)            = o0;
      *(v4f*)(out + NPIX + p4)     = o1;
      *(v4f*)(out + 2 * NPIX + p4) = o2;
    }
  }
}

extern "C" void kernel_launch(void* const* d_in, const int* in_sizes, int n_in,
                              void* d_out, int out_size, void* d_ws, size_t ws_size,
                              hipStream_t stream) {
  const float* tgt = (const float*)d_in[0];   // "input"
  const float* src = (const float*)d_in[1];   // "source_tensor"
  float* out = (float*)d_out;
  float* ws  = (float*)d_ws;   // needs (NB_RED*18 + 12) floats ≈ 18.5 KB

  mh_reduce   <<<NB_RED, NT_RED, 0, stream>>>(tgt, src, ws);
  mh_finalize <<<1, 32, 0, stream>>>(ws);
  mh_apply_wmma<<<1024, 256, 0, stream>>>(tgt, out, ws + WS_RESULT_OFF);
}